// GIN_22376779612621
// MI455X (gfx1250) — compile-verified
//
#include <hip/hip_runtime.h>

// GIN (3-layer, H=64) for MI455X gfx1250.
// Pipeline per call:
//   for l in 0..2:  zero(agg); scatter(agg += h[src] at dst);  h = mlp(h+agg, W1,b1,W2,b2)
//   zero(pooled); pool(pooled[batch[n]] += h[n]); readout -> out[G]
// Workspace: h_buf[N*64] | agg[N*64] | pooled[G*64]  (~51.5 MB floats)

#define HN 64
#define SXS 68   // padded LDS row stride (floats): stride%64==4 -> conflict-free column reads

typedef __attribute__((ext_vector_type(2))) float v2f;
typedef __attribute__((ext_vector_type(8))) float v8f;
typedef int v4i __attribute__((vector_size(4 * sizeof(int))));

// gfx1250 async global->LDS copies (ASYNCcnt) if the toolchain exposes them.
#if defined(__has_builtin)
# if __has_builtin(__builtin_amdgcn_global_load_async_to_lds_b128) && \
     __has_builtin(__builtin_amdgcn_s_wait_asynccnt)
#  define GIN_ASYNC_LDS 1
# endif
#endif
#ifndef GIN_ASYNC_LDS
# define GIN_ASYNC_LDS 0
#endif

__global__ void gin_zero_kernel(float* __restrict__ p, int n4) {
    int i = blockIdx.x * blockDim.x + threadIdx.x;
    if (i < n4) ((float4*)p)[i] = make_float4(0.f, 0.f, 0.f, 0.f);
}

// agg[dst[e]] += h[src[e]] : 16 lanes per edge, float4 per lane (128-bit gather)
__global__ void gin_scatter_kernel(const float* __restrict__ h, const int* __restrict__ src,
                                   const int* __restrict__ dst, float* __restrict__ agg, int E) {
    int t = blockIdx.x * blockDim.x + threadIdx.x;
    int e = t >> 4;
    if (e >= E) return;
    int f = (t & 15) << 2;
    int s = src[e], d = dst[e];
    const float4 v = *(const float4*)(h + (size_t)s * HN + f);
    float* o = agg + (size_t)d * HN + f;
    atomicAdd(o + 0, v.x);
    atomicAdd(o + 1, v.y);
    atomicAdd(o + 2, v.z);
    atomicAdd(o + 3, v.w);
}

// Issue 64x64 f32 weight matrix into LDS staging (raw row-major, 4096 floats).
__device__ __forceinline__ void gin_load_w_raw(const float* __restrict__ W, float* stage,
                                               int tid) {
#if GIN_ASYNC_LDS
    for (int i = tid; i < 1024; i += 256) {
        __builtin_amdgcn_global_load_async_to_lds_b128(
            (__attribute__((address_space(1))) v4i*)(W + i * 4),
            (__attribute__((address_space(3))) v4i*)(stage + i * 4), 0, 0);
    }
    __builtin_amdgcn_s_wait_asynccnt(0);
#else
    for (int i = tid; i < 1024; i += 256)
        ((float4*)stage)[i] = ((const float4*)W)[i];
#endif
}

// Reformat raw W (row-major 64x64) into row-pair-interleaved sWP:
//   sWP[(p*64 + c)*2 + r] = W[2p + r][c]
// so a B fragment (rows {2h,2h+1} of a K-quad at column c) is one contiguous b64.
__device__ __forceinline__ void gin_reformat_w(const float* stage, float* sWP, int tid) {
    for (int i = tid; i < 1024; i += 256) {
        const int p  = i >> 5;          // pair 0..31 (rows 2p, 2p+1)
        const int cc = (i & 31) << 1;   // column pair
        const float2 r0 = *(const float2*)(stage + (2 * p) * 64 + cc);
        const float2 r1 = *(const float2*)(stage + (2 * p + 1) * 64 + cc);
        *(float4*)(sWP + (((p << 6) + cc) << 1)) = make_float4(r0.x, r1.x, r0.y, r1.y);
    }
}

// Fused 2-layer MLP on a 64-row block:
//   hout = relu( relu( (hin+agg) @ W1 + b1 ) @ W2 + b2 )
// 256 threads = 8 wave32; each wave computes two 16x16 output tiles per stage
// via v_wmma_f32_16x16x4_f32 chained over K=64 (16 steps).
__global__ __launch_bounds__(256)
void gin_mlp_kernel(const float* __restrict__ hin, const float* __restrict__ agg,
                    const float* __restrict__ W1, const float* __restrict__ b1,
                    const float* __restrict__ W2, const float* __restrict__ b2,
                    float* __restrict__ hout, int N) {
    __shared__ float sX[64 * SXS];   // X tile; first 4096 floats reused as W2 raw staging
    __shared__ float sZ[64 * SXS];   // Z tile; first 4096 floats reused as W1 raw staging
    __shared__ float sWP[64 * 64];   // row-pair-interleaved weights
    __shared__ float sB[64];

    const int tid  = threadIdx.x;
    const int r0   = blockIdx.x * 64;
    const int wave = tid >> 5;
    const int lane = tid & 31;
    const int half = lane >> 4;   // 0: lanes 0-15, 1: lanes 16-31
    const int l16  = lane & 15;

    // Pull stage-2 weights toward the caches early (global_prefetch_b8).
    __builtin_prefetch(W2 + (tid << 4), 0, 1);

    // (1) Kick off W1 -> LDS raw staging (async when available), bias sync.
    gin_load_w_raw(W1, sZ, tid);
    if (tid < 64) sB[tid] = b1[tid];

    // (2) Stage X = hin + agg for this 64-row block (zero-pad rows >= N).
    {
        int row  = tid >> 2;
        int c0   = (tid & 3) << 4;
        int grow = r0 + row;
        for (int q = 0; q < 4; ++q) {
            float4 v = make_float4(0.f, 0.f, 0.f, 0.f);
            if (grow < N) {
                const float4 a = *(const float4*)(hin + (size_t)grow * HN + c0 + q * 4);
                const float4 g = *(const float4*)(agg + (size_t)grow * HN + c0 + q * 4);
                v = make_float4(a.x + g.x, a.y + g.y, a.z + g.z, a.w + g.w);
            }
            *(float4*)(sX + row * SXS + c0 + q * 4) = v;
        }
    }
    __syncthreads();

    // (3) Reformat W1 into pair-interleaved layout.
    gin_reformat_w(sZ, sWP, tid);
    __syncthreads();

    // ---- Stage 1: Z = relu(X @ W1 + b1) -> LDS ----
    for (int t2 = 0; t2 < 2; ++t2) {
        const int tile = wave * 2 + t2;       // 0..15
        const int mr = (tile >> 2) << 4;      // row-tile base
        const int nc = (tile & 3) << 4;       // col-tile base
        v8f acc = {};
        const float* ax = sX + (mr + l16) * SXS + 2 * half;       // A: lane m, K={2h,2h+1}
        const float* bx = sWP + (((half << 6) + nc + l16) << 1);  // B: pair (2kk+h), col nc+l
#pragma unroll
        for (int kk = 0; kk < 16; ++kk) {
            v2f a, b;
            a.x = ax[kk * 4 + 0];
            a.y = ax[kk * 4 + 1];
            b   = *(const v2f*)(bx + kk * 256);   // one ds_load_b64, pre-paired
            acc = __builtin_amdgcn_wmma_f32_16x16x4_f32(false, a, false, b,
                                                        (short)0, acc, false, false);
        }
        const float bias = sB[nc + l16];
#pragma unroll
        for (int i = 0; i < 8; ++i) {
            // C/D layout: VGPR i -> row mr + i (+8 for lanes 16-31), col nc + l16
            sZ[(mr + 8 * half + i) * SXS + nc + l16] = fmaxf(acc[i] + bias, 0.f);
        }
    }
    __syncthreads();

    // (4) W2 -> raw staging in (now dead) sX region, then reformat.
    gin_load_w_raw(W2, sX, tid);
    if (tid < 64) sB[tid] = b2[tid];
    __syncthreads();
    gin_reformat_w(sX, sWP, tid);
    __syncthreads();

    // ---- Stage 2: H' = relu(Z @ W2 + b2) -> global ----
    for (int t2 = 0; t2 < 2; ++t2) {
        const int tile = wave * 2 + t2;
        const int mr = (tile >> 2) << 4;
        const int nc = (tile & 3) << 4;
        v8f acc = {};
        const float* ax = sZ + (mr + l16) * SXS + 2 * half;
        const float* bx = sWP + (((half << 6) + nc + l16) << 1);
#pragma unroll
        for (int kk = 0; kk < 16; ++kk) {
            v2f a, b;
            a.x = ax[kk * 4 + 0];
            a.y = ax[kk * 4 + 1];
            b   = *(const v2f*)(bx + kk * 256);
            acc = __builtin_amdgcn_wmma_f32_16x16x4_f32(false, a, false, b,
                                                        (short)0, acc, false, false);
        }
        const float bias = sB[nc + l16];
#pragma unroll
        for (int i = 0; i < 8; ++i) {
            const int grow = r0 + mr + 8 * half + i;
            if (grow < N) hout[(size_t)grow * HN + nc + l16] = fmaxf(acc[i] + bias, 0.f);
        }
    }
}

// pooled[batch[n]] += h[n] : 16 lanes per node, float4 per lane
__global__ void gin_pool_kernel(const float* __restrict__ h, const int* __restrict__ batch,
                                float* __restrict__ pooled, int N) {
    int t = blockIdx.x * blockDim.x + threadIdx.x;
    int n = t >> 4;
    if (n >= N) return;
    int f = (t & 15) << 2;
    int g = batch[n];
    const float4 v = *(const float4*)(h + (size_t)n * HN + f);
    float* o = pooled + (size_t)g * HN + f;
    atomicAdd(o + 0, v.x);
    atomicAdd(o + 1, v.y);
    atomicAdd(o + 2, v.z);
    atomicAdd(o + 3, v.w);
}

// out[g] = relu(pooled[g] @ Wr1 + br1) @ Wr2 + br2 ; one wave32 per graph
__global__ void gin_readout_kernel(const float* __restrict__ pooled,
                                   const float* __restrict__ Wr1, const float* __restrict__ br1,
                                   const float* __restrict__ Wr2, const float* __restrict__ br2,
                                   float* __restrict__ out, int G) {
    int g = blockIdx.x * (blockDim.x >> 5) + (threadIdx.x >> 5);
    int lane = threadIdx.x & 31;
    if (g >= G) return;
    float acc = 0.f;
#pragma unroll
    for (int jj = 0; jj < 2; ++jj) {
        int j = lane + jj * 32;
        float t = br1[j];
        for (int k = 0; k < HN; ++k) t = fmaf(pooled[(size_t)g * HN + k], Wr1[k * HN + j], t);
        acc = fmaf(fmaxf(t, 0.f), Wr2[j], acc);
    }
    for (int off = 16; off > 0; off >>= 1) acc += __shfl_down(acc, off, 32);
    if (lane == 0) out[g] = acc + br2[0];
}

extern "C" void kernel_launch(void* const* d_in, const int* in_sizes, int n_in,
                              void* d_out, int out_size, void* d_ws, size_t ws_size,
                              hipStream_t stream) {
    const float* x   = (const float*)d_in[0];
    const int*   ei  = (const int*)d_in[1];
    const int*   bi  = (const int*)d_in[2];
    const float* W1  = (const float*)d_in[3];
    const float* b1  = (const float*)d_in[4];
    const float* W2  = (const float*)d_in[5];
    const float* b2  = (const float*)d_in[6];
    const float* Wr1 = (const float*)d_in[7];
    const float* br1 = (const float*)d_in[8];
    const float* Wr2 = (const float*)d_in[9];
    const float* br2 = (const float*)d_in[10];
    float* out = (float*)d_out;

    const int N = in_sizes[0] / HN;      // 100000
    const int E = in_sizes[1] / 2;       // 1250000
    const int L = in_sizes[4] / HN;      // 3
    const int G = out_size;              // 1024 (T == 1)

    float* h_buf  = (float*)d_ws;
    float* agg    = h_buf + (size_t)N * HN;
    float* pooled = agg + (size_t)N * HN;

    const int aggN4  = (N * HN) / 4;
    const int poolN4 = (G * HN) / 4;
    const int nMlpBlk = (N + 63) / 64;
    const int nScatBlk = (E * 16 + 255) / 256;
    const int nPoolBlk = (N * 16 + 255) / 256;

    for (int l = 0; l < L; ++l) {
        const float* hcur = (l == 0) ? x : h_buf;
        gin_zero_kernel<<<(aggN4 + 255) / 256, 256, 0, stream>>>(agg, aggN4);
        gin_scatter_kernel<<<nScatBlk, 256, 0, stream>>>(hcur, ei, ei + E, agg, E);
        gin_mlp_kernel<<<nMlpBlk, 256, 0, stream>>>(hcur, agg,
                                                    W1 + (size_t)l * HN * HN, b1 + (size_t)l * HN,
                                                    W2 + (size_t)l * HN * HN, b2 + (size_t)l * HN,
                                                    h_buf, N);
    }
    gin_zero_kernel<<<(poolN4 + 255) / 256, 256, 0, stream>>>(pooled, poolN4);
    gin_pool_kernel<<<nPoolBlk, 256, 0, stream>>>(h_buf, bi, pooled, N);
    gin_readout_kernel<<<(G + 7) / 8, 256, 0, stream>>>(pooled, Wr1, br1, Wr2, br2, out, G);
}